// Code2InvMulti_78005196030028
// MI455X (gfx1250) — compile-verified
//
#include <hip/hip_runtime.h>
#include <math.h>

// ---------------------------------------------------------------------------
// Code2InvMulti GNN forward for MI455X (gfx1250, wave32).
//
// Phases per level: fp32 WMMA GEMM (conv) -> zero msg -> edge scatter-add
// (fp32 global atomics, L2-resident slabs) -> tanh -> fp32 WMMA GEMM
// (merge, fused bias+residual+tanh).  Scatter is L2-bandwidth bound, so fp32
// V_WMMA_F32_16X16X4_F32 is used for full reference precision (GEMM is not
// the bottleneck at 23.3 TB/s HBM / 192 MB L2).
// ---------------------------------------------------------------------------

#define D_LAT   128
#define ETYPES  4
#define ED      512   // ETYPES * D_LAT

typedef float v2f __attribute__((ext_vector_type(2)));
typedef float v8f __attribute__((ext_vector_type(8)));

__device__ __forceinline__ v8f wmma_f32_16x16x4(v2f a, v2f b, v8f c) {
  // (neg_a, A, neg_b, B, c_mod, C, reuse_a, reuse_b)
  return __builtin_amdgcn_wmma_f32_16x16x4_f32(false, a, false, b, (short)0, c,
                                               false, false);
}

// out[m,n] = act( sum_k A[m,k] * W[k,n] + bias[n] + (R ? R[m,n] : 0) )
// A: [nRows,K] row-major, W: [K,nCols] row-major.
// Block = 128 threads (4 waves); each wave computes a 32x64 tile as
// 2 (M) x 4 (N) fragments of V_WMMA_F32_16X16X4_F32.
__global__ __launch_bounds__(128)
void gemm_bias_act(const float* __restrict__ A, const float* __restrict__ W,
                   const float* __restrict__ bias, const float* __restrict__ R,
                   float* __restrict__ out, int nRows, int K, int nCols,
                   int doTanh) {
  const int lane = threadIdx.x & 31;
  const int wave = threadIdx.x >> 5;
  const int lm   = lane & 15;          // M (for A) / N (for B,C) within tile
  const int kk   = (lane >> 4) << 1;   // K pair select: lanes 0-15 -> K0,K1; 16-31 -> K2,K3
  const int mo   = (lane >> 4) << 3;   // C/D: lanes 16-31 hold M = r + 8

  const int rowBase = blockIdx.x * 128 + wave * 32;
  if (rowBase >= nRows) return;        // nRows is a multiple of 32
  const int colBase = blockIdx.y * 64;

  // Accumulators initialized with bias (bias depends only on N -> same value
  // in all 8 C-fragment registers of a lane).
  v8f acc[2][4];
#pragma unroll
  for (int nt = 0; nt < 4; ++nt) {
    const float bv = bias[colBase + nt * 16 + lm];
#pragma unroll
    for (int mt = 0; mt < 2; ++mt)
#pragma unroll
      for (int r = 0; r < 8; ++r) acc[mt][nt][r] = bv;
  }

  // A fragment rows (clamped for safety; EXEC stays all-1s for WMMA).
  const int r0 = min(rowBase + lm,      nRows - 1);
  const int r1 = min(rowBase + 16 + lm, nRows - 1);
  const float* a0p = A + (size_t)r0 * K + kk;   // 8B-aligned -> b64 loads
  const float* a1p = A + (size_t)r1 * K + kk;
  const float* wp  = W + colBase + lm;

  for (int k = 0; k < K; k += 4) {
    v2f a[2];
    a[0] = *(const v2f*)(a0p + k);
    a[1] = *(const v2f*)(a1p + k);

    const float* w0 = wp + (size_t)(k + kk) * nCols;
    const float* w1 = w0 + nCols;
    v2f b[4];
#pragma unroll
    for (int nt = 0; nt < 4; ++nt) {
      b[nt].x = w0[nt * 16];
      b[nt].y = w1[nt * 16];
    }
#pragma unroll
    for (int mt = 0; mt < 2; ++mt)
#pragma unroll
      for (int nt = 0; nt < 4; ++nt)
        acc[mt][nt] = wmma_f32_16x16x4(a[mt], b[nt], acc[mt][nt]);
  }

  // Epilogue: residual + tanh + store.  C/D layout: VGPR r, lanes 0-15 ->
  // M=r, lanes 16-31 -> M=r+8; N = lane%16.
#pragma unroll
  for (int mt = 0; mt < 2; ++mt)
#pragma unroll
    for (int nt = 0; nt < 4; ++nt) {
      const int n = colBase + nt * 16 + lm;
#pragma unroll
      for (int r = 0; r < 8; ++r) {
        const int m = rowBase + mt * 16 + mo + r;
        if (m < nRows) {
          float v = acc[mt][nt][r];
          if (R) v += R[(size_t)m * nCols + n];
          if (doTanh) v = tanhf(v);
          out[(size_t)m * nCols + n] = v;
        }
      }
    }
}

// One wave per edge (grid-stride); lane j moves float4 at element offset 4j.
// blockIdx.y = edge type.
__global__ __launch_bounds__(256)
void scatter_add(const float* __restrict__ conv, const int* __restrict__ src,
                 const int* __restrict__ dst, float* __restrict__ msg,
                 int nEdges) {
  const int lane   = threadIdx.x & 31;
  const int e      = blockIdx.y;
  int       wv     = (int)((blockIdx.x * blockDim.x + threadIdx.x) >> 5);
  const int nWaves = (int)((gridDim.x * blockDim.x) >> 5);

  const int*   srcE = src + (size_t)e * nEdges;
  const int*   dstE = dst + (size_t)e * nEdges;
  const size_t co   = (size_t)e * D_LAT + (size_t)lane * 4;

  for (int i = wv; i < nEdges; i += nWaves) {
    const int s = srcE[i];
    const int d = dstE[i];
    const float4 v = *(const float4*)(conv + (size_t)s * ED + co);  // b128
    float* p = msg + (size_t)d * ED + co;
    unsafeAtomicAdd(p + 0, v.x);   // -> global_atomic_add_f32 (no return)
    unsafeAtomicAdd(p + 1, v.y);
    unsafeAtomicAdd(p + 2, v.z);
    unsafeAtomicAdd(p + 3, v.w);
  }
}

__global__ __launch_bounds__(256)
void zero4(float4* __restrict__ p, size_t n4) {
  size_t i = (size_t)blockIdx.x * blockDim.x + threadIdx.x;
  const size_t stride = (size_t)gridDim.x * blockDim.x;
  const float4 z = make_float4(0.f, 0.f, 0.f, 0.f);
  for (; i < n4; i += stride) p[i] = z;
}

__global__ __launch_bounds__(256)
void tanh4(float4* __restrict__ p, size_t n4) {
  size_t i = (size_t)blockIdx.x * blockDim.x + threadIdx.x;
  const size_t stride = (size_t)gridDim.x * blockDim.x;
  for (; i < n4; i += stride) {
    float4 v = p[i];
    v.x = tanhf(v.x); v.y = tanhf(v.y); v.z = tanhf(v.z); v.w = tanhf(v.w);
    p[i] = v;
  }
}

extern "C" void kernel_launch(void* const* d_in, const int* in_sizes, int n_in,
                              void* d_out, int out_size, void* d_ws,
                              size_t ws_size, hipStream_t stream) {
  const float* node_feat = (const float*)d_in[0];
  const int*   edge_src  = (const int*)d_in[1];
  const int*   edge_dst  = (const int*)d_in[2];
  const float* w_n2l     = (const float*)d_in[3];
  const float* b_n2l     = (const float*)d_in[4];
  const float* conv_w    = (const float*)d_in[5];
  const float* conv_b    = (const float*)d_in[6];
  const float* merge_w   = (const float*)d_in[7];
  const float* merge_b   = (const float*)d_in[8];

  const int N      = in_sizes[0] / D_LAT;   // 100000 (multiple of 32)
  const int nEdges = in_sizes[1] / ETYPES;  // edges per type

  float* h    = (float*)d_out;              // [N, 128], evolves across levels
  float* conv = (float*)d_ws;               // [N, 512]
  float* msg  = conv + (size_t)N * ED;      // [N, 512]

  const dim3 blk(128);
  const int  rowTiles = (N + 127) / 128;
  const size_t n4 = (size_t)N * ED / 4;

  // h = tanh(node_feat @ w_n2l + b_n2l)
  gemm_bias_act<<<dim3(rowTiles, D_LAT / 64), blk, 0, stream>>>(
      node_feat, w_n2l, b_n2l, nullptr, h, N, D_LAT, D_LAT, 1);

  for (int lv = 0; lv < 3; ++lv) {
    // conv = h @ conv_w[lv] + conv_b[lv]   (linear, no activation)
    gemm_bias_act<<<dim3(rowTiles, ED / 64), blk, 0, stream>>>(
        h, conv_w + (size_t)lv * D_LAT * ED, conv_b + (size_t)lv * ED, nullptr,
        conv, N, D_LAT, ED, 0);

    zero4<<<2048, 256, 0, stream>>>((float4*)msg, n4);

    // per-type gather + scatter-add into msg[N, E*D]
    scatter_add<<<dim3(4096, ETYPES), 256, 0, stream>>>(conv, edge_src,
                                                        edge_dst, msg, nEdges);

    tanh4<<<2048, 256, 0, stream>>>((float4*)msg, n4);

    // h = tanh(msg @ merge_w[lv] + merge_b[lv] + h)   (in-place residual safe)
    gemm_bias_act<<<dim3(rowTiles, D_LAT / 64), blk, 0, stream>>>(
        msg, merge_w + (size_t)lv * ED * D_LAT, merge_b + (size_t)lv * D_LAT,
        h, h, N, ED, D_LAT, 1);
  }
}